// FusingCrossAttentionV2_43817256354004
// MI455X (gfx1250) — compile-verified
//
#include <hip/hip_runtime.h>

// ---------------------------------------------------------------------------
// Fused BEV deformable cross-attention for MI455X (gfx1250, wave32, WMMA)
//   B=2, N=200*200=40000, C=128, HEADS=8 (dh=16), PTS=4
// Pipeline (6 kernels on `stream`):
//   0) pack  W_off|W_att -> Wcat[128x96], b_off|b_att -> bcat[96]
//   1) q96   = (query+query_pos) @ Wcat + bcat                    (WMMA f32)
//   2) in-place: q96 -> sampling coords (gx,gy) + softmax attention
//   3) value = input_feats @ W_val + b_val                        (WMMA f32)
//   4) mid   = sum_p attn * bilinear(value)                       (gather)
//   5) out   = mid @ W_out + b_out + query                        (WMMA f32)
// ---------------------------------------------------------------------------

typedef float v2f __attribute__((ext_vector_type(2)));
typedef float v8f __attribute__((ext_vector_type(8)));

#define BEV_H   200
#define BEV_W   200
#define NQ      (BEV_H * BEV_W)   // 40000
#define NB      2
#define CDIM    128
#define NHEADS  8
#define NPTS    4
#define NROWS   (NB * NQ)         // 80000 total query rows

// ---------------------------------------------------------------------------
// K=128 WMMA GEMM:  out[row, 0:NCOLS] = (A0 (+A1)) @ W + bias (+ resid)
// Block: 256 threads = 8 wave32; 64-row A slab in LDS (stride 132: the
// fragment ds_load_b64 of lane l hits dword banks {4*l15+2*hi, +1} -> all 64
// banks distinct).  Wave w: row strip (w>>1)*16, column half (w&1)*(NCOLS/2),
// NT = NCOLS/32 16x16 accum tiles.  K loop fully unrolled: B-fragment loads
// are global_load_b32 with immediate offsets (constexpr stride), A-fragment
// loads are ds_load_b64 with immediate offsets.
// ---------------------------------------------------------------------------
template <int NCOLS, bool ADD_A1, bool ADD_RES>
__global__ __launch_bounds__(256)
void wmma_gemm(const float* __restrict__ A0, const float* __restrict__ A1,
               const float* __restrict__ W,  const float* __restrict__ bias,
               const float* __restrict__ resid, float* __restrict__ out) {
  constexpr int AROWS = 64;
  constexpr int LDA   = 132;          // pad: conflict-free b64 fragment reads
  constexpr int NT    = NCOLS / 32;   // 16-col tiles per wave

  __shared__ float sA[AROWS * LDA];

  const int t    = threadIdx.x;
  const int lane = t & 31;            // wave32
  const int wave = t >> 5;            // 0..7
  const int hi   = lane >> 4;         // half-wave: carries K pairs / M+8
  const int l15  = lane & 15;
  const int rowStrip = wave >> 1;     // 0..3
  const int chalf    = wave & 1;      // 0..1
  const int rowBase  = blockIdx.x * AROWS;

  // warm WGP$/L2 with the weight matrix (global_prefetch_b8; 1B/lane spans W)
  __builtin_prefetch(W + (size_t)t * ((NCOLS * CDIM) / 256), 0, 0);

  // ---- cooperative load of the 64x128 A slab (optionally fused add) ----
  for (int i = t; i < AROWS * (CDIM / 4); i += 256) {
    const int r  = i >> 5;            // 32 float4 per row
    const int c4 = i & 31;
    float4 v = *(const float4*)(A0 + (size_t)(rowBase + r) * CDIM + c4 * 4);
    if constexpr (ADD_A1) {
      const float4 w = *(const float4*)(A1 + (size_t)(rowBase + r) * CDIM + c4 * 4);
      v.x += w.x; v.y += w.y; v.z += w.z; v.w += w.w;
    }
    float* dst = &sA[r * LDA + c4 * 4];
    dst[0] = v.x; dst[1] = v.y; dst[2] = v.z; dst[3] = v.w;
  }
  __syncthreads();

  v8f acc[NT];
#pragma unroll
  for (int tt = 0; tt < NT; ++tt)
#pragma unroll
    for (int j = 0; j < 8; ++j) acc[tt][j] = 0.0f;

  // A fragment base: row = rowStrip*16 + l15, K pair selected by hi
  const float* aPtr = &sA[(rowStrip * 16 + l15) * LDA + 2 * hi];

  // per-tile weight column base (constexpr stride NCOLS -> immediate offsets)
  const float* wPtr[NT];
  int          ncol[NT];
  float        bv[NT];
#pragma unroll
  for (int tt = 0; tt < NT; ++tt) {
    const int n = chalf * (NCOLS / 2) + tt * 16 + l15;
    ncol[tt] = n;
    bv[tt]   = bias[n];
    wPtr[tt] = W + (size_t)(2 * hi) * NCOLS + n;
  }

  // ---- fully unrolled K loop: 32 steps of K=4, NT wmma per step ----
#pragma unroll
  for (int ks = 0; ks < CDIM / 4; ++ks) {
    const v2f af = { aPtr[ks * 4], aPtr[ks * 4 + 1] };
#pragma unroll
    for (int tt = 0; tt < NT; ++tt) {
      const v2f bf = { wPtr[tt][(size_t)(ks * 4) * NCOLS],
                       wPtr[tt][(size_t)(ks * 4 + 1) * NCOLS] };
      acc[tt] = __builtin_amdgcn_wmma_f32_16x16x4_f32(
          /*neg_a=*/false, af, /*neg_b=*/false, bf,
          /*c_mod=*/(short)0, acc[tt], /*reuse_a=*/false, /*reuse_b=*/false);
    }
  }

  // ---- epilogue: C/D layout VGPR r -> M = r + 8*hi, lane&15 -> N ----
#pragma unroll
  for (int tt = 0; tt < NT; ++tt) {
    const int n = ncol[tt];
#pragma unroll
    for (int r = 0; r < 8; ++r) {
      const size_t row = (size_t)rowBase + rowStrip * 16 + 8 * hi + r;
      float v = acc[tt][r] + bv[tt];
      if constexpr (ADD_RES) v += resid[row * CDIM + n];
      out[row * NCOLS + n] = v;
    }
  }
}

// ---------------------------------------------------------------------------
// Pack [W_off | W_att] -> Wcat[128x96], [b_off | b_att] -> bcat[96]
// ---------------------------------------------------------------------------
__global__ __launch_bounds__(256)
void pack_wcat(const float* __restrict__ W_off, const float* __restrict__ b_off,
               const float* __restrict__ W_att, const float* __restrict__ b_att,
               float* __restrict__ Wcat, float* __restrict__ bcat) {
  const int idx = blockIdx.x * 256 + threadIdx.x;
  if (idx < CDIM * 96) {
    const int k = idx / 96;
    const int n = idx - k * 96;
    Wcat[idx] = (n < 64) ? W_off[k * 64 + n] : W_att[k * 32 + (n - 64)];
  }
  if (idx < 96) bcat[idx] = (idx < 64) ? b_off[idx] : b_att[idx - 64];
}

// ---------------------------------------------------------------------------
// In-place: q96[row, 0:96] (8 offsets + 4 logits per head) ->
//           (gx,gy) sample coords + softmax attention weights.
// Reference math:  gx = (iz+0.5)*W/H + off0 - 0.5 ; gy = (ix+0.5)*H/W + off1 - 0.5
// (the torch module stacks ref as (z,x), grid_sample reads x first)
// ---------------------------------------------------------------------------
__global__ __launch_bounds__(256)
void prep_loc_kernel(float* __restrict__ q96) {
  const long i = (long)blockIdx.x * blockDim.x + threadIdx.x;  // over NROWS*NHEADS
  if (i >= (long)NROWS * NHEADS) return;
  const int  h   = (int)(i & (NHEADS - 1));
  const long row = i >> 3;
  const int  n   = (int)(row % NQ);
  const int  iz  = n / BEV_W;
  const int  ix  = n - iz * BEV_W;

  float* base = q96 + row * 96;
  float offv[2 * NPTS], att[NPTS];
#pragma unroll
  for (int p = 0; p < NPTS; ++p) {
    offv[2 * p]     = base[h * 8 + 2 * p];
    offv[2 * p + 1] = base[h * 8 + 2 * p + 1];
    att[p]          = base[64 + h * 4 + p];
  }
  float m = att[0];
#pragma unroll
  for (int p = 1; p < NPTS; ++p) m = fmaxf(m, att[p]);
  float e[NPTS], s = 0.0f;
#pragma unroll
  for (int p = 0; p < NPTS; ++p) { e[p] = __expf(att[p] - m); s += e[p]; }
  const float inv = 1.0f / s;

  const float sx = (float)BEV_W / (float)BEV_H;
  const float sy = (float)BEV_H / (float)BEV_W;
#pragma unroll
  for (int p = 0; p < NPTS; ++p) {
    base[h * 8 + 2 * p]     = ((float)iz + 0.5f) * sx + offv[2 * p]     - 0.5f;  // gx
    base[h * 8 + 2 * p + 1] = ((float)ix + 0.5f) * sy + offv[2 * p + 1] - 0.5f;  // gy
    base[64 + h * 4 + p]    = e[p] * inv;                                        // attn
  }
}

// ---------------------------------------------------------------------------
// Bilinear gather + attention reduce: mid[row, c] = sum_p attn * tap4(value)
// Block = 256 threads = 2 query rows; thread c: head = c>>4, dim = c&15.
// Per-block loc/attn floats staged once in LDS (192 floats) instead of 12
// broadcast global loads per thread.  16 consecutive lanes read 16
// consecutive floats of a value row -> 64B segments; value (41 MB) stays
// resident in the 192 MB L2.
// ---------------------------------------------------------------------------
__global__ __launch_bounds__(256)
void sample_kernel(const float* __restrict__ loc96, const float* __restrict__ value,
                   float* __restrict__ mid) {
  __shared__ float sLoc[2 * 96];
  const long row0 = (long)blockIdx.x * 2;
  if (threadIdx.x < 192) sLoc[threadIdx.x] = loc96[row0 * 96 + threadIdx.x];
  __syncthreads();

  const int  rl  = threadIdx.x >> 7;       // 0..1 : row within block
  const long row = row0 + rl;
  const int  c   = threadIdx.x & 127;
  const int  h   = c >> 4;
  const int  b   = (int)(row / NQ);

  const float* base = sLoc + rl * 96;
  const float* vb   = value + (size_t)b * NQ * CDIM;

  float acc = 0.0f;
#pragma unroll
  for (int p = 0; p < NPTS; ++p) {
    const float gx = base[h * 8 + 2 * p];
    const float gy = base[h * 8 + 2 * p + 1];
    const float aw = base[64 + h * 4 + p];
    const float x0f = floorf(gx), y0f = floorf(gy);
    const int   x0 = (int)x0f, y0 = (int)y0f;
    const float wx1 = gx - x0f, wy1 = gy - y0f;
    const float wx0 = 1.0f - wx1, wy0 = 1.0f - wy1;

    float s = 0.0f;
    if (y0 >= 0 && y0 < BEV_H && x0 >= 0 && x0 < BEV_W)
      s += wy0 * wx0 * vb[((size_t)y0 * BEV_W + x0) * CDIM + c];
    if (y0 >= 0 && y0 < BEV_H && x0 + 1 >= 0 && x0 + 1 < BEV_W)
      s += wy0 * wx1 * vb[((size_t)y0 * BEV_W + x0 + 1) * CDIM + c];
    if (y0 + 1 >= 0 && y0 + 1 < BEV_H && x0 >= 0 && x0 < BEV_W)
      s += wy1 * wx0 * vb[((size_t)(y0 + 1) * BEV_W + x0) * CDIM + c];
    if (y0 + 1 >= 0 && y0 + 1 < BEV_H && x0 + 1 >= 0 && x0 + 1 < BEV_W)
      s += wy1 * wx1 * vb[((size_t)(y0 + 1) * BEV_W + x0 + 1) * CDIM + c];

    acc += aw * s;
  }
  mid[row * CDIM + c] = acc;
}

// ---------------------------------------------------------------------------
extern "C" void kernel_launch(void* const* d_in, const int* in_sizes, int n_in,
                              void* d_out, int out_size, void* d_ws, size_t ws_size,
                              hipStream_t stream) {
  const float* query       = (const float*)d_in[0];
  const float* input_feats = (const float*)d_in[1];
  const float* query_pos   = (const float*)d_in[2];
  const float* W_off       = (const float*)d_in[3];
  const float* b_off       = (const float*)d_in[4];
  const float* W_att       = (const float*)d_in[5];
  const float* b_att       = (const float*)d_in[6];
  const float* W_val       = (const float*)d_in[7];
  const float* b_val       = (const float*)d_in[8];
  const float* W_out       = (const float*)d_in[9];
  const float* b_out       = (const float*)d_in[10];
  float*       out         = (float*)d_out;

  float* q96  = (float*)d_ws;                        // NROWS*96 (reused in-place)
  float* val  = q96  + (size_t)NROWS * 96;           // NROWS*128
  float* mid  = val  + (size_t)NROWS * CDIM;         // NROWS*128
  float* Wcat = mid  + (size_t)NROWS * CDIM;         // 128*96
  float* bcat = Wcat + (size_t)CDIM * 96;            // 96

  const dim3 blk(256);
  const int  gemmBlocks = NROWS / 64;                // 80000/64 = 1250 exact

  // 0) pack concatenated offset/attention weights
  pack_wcat<<<(CDIM * 96 + 255) / 256, blk, 0, stream>>>(
      W_off, b_off, W_att, b_att, Wcat, bcat);

  // 1) offsets+logits GEMM (fused q = query + query_pos)
  wmma_gemm<96, true, false><<<gemmBlocks, blk, 0, stream>>>(
      query, query_pos, Wcat, bcat, nullptr, q96);

  // 2) coords + softmax, in place
  prep_loc_kernel<<<(NROWS * NHEADS + 255) / 256, blk, 0, stream>>>(q96);

  // 3) value projection GEMM
  wmma_gemm<128, false, false><<<gemmBlocks, blk, 0, stream>>>(
      input_feats, nullptr, W_val, b_val, nullptr, val);

  // 4) bilinear sampling + attention reduce
  sample_kernel<<<NROWS / 2, blk, 0, stream>>>(q96, val, mid);

  // 5) output projection + bias + residual
  wmma_gemm<128, false, true><<<gemmBlocks, blk, 0, stream>>>(
      mid, nullptr, W_out, b_out, query, out);
}